// ChainCRF_44710609551703
// MI455X (gfx1250) — compile-verified
//
#include <hip/hip_runtime.h>
#include <stdint.h>

#define NL     65     // num labels incl pad
#define NSTEP  512
#define NBATCH 32
#define P      4      // row partitions per column reduction
#define DEPTH  6      // TDM pipeline depth (6 x 16.9KB tiles in flight)
#define ACT    (P * NL)   // 260 active compute threads
#define TPB    288        // 9 wave32 waves

typedef unsigned int       u32;
typedef unsigned long long u64;
typedef __attribute__((ext_vector_type(4))) u32 v4u;
typedef __attribute__((ext_vector_type(4))) int v4i;
typedef __attribute__((ext_vector_type(8))) int v8i;

__device__ __forceinline__ float fast_exp(float x) {
    return __builtin_amdgcn_exp2f(x * 1.44269504088896340736f);   // v_exp_f32
}
__device__ __forceinline__ float fast_log(float x) {
    return __builtin_amdgcn_logf(x) * 0.693147180559945309417f;   // v_log_f32
}

// Issue one TDM DMA: contiguous 1-D tile of `tile_elems` f32 from global -> LDS.
// D# per CDNA5 ISA ch.8: group0 = {count|flags, lds_addr, global_addr, type=2},
// group1 = {data_size=4B, tensor_dim0, tensor_dim1=1, tile_dim0, strides}.
// tile_elems == 0 => NOP descriptor (still completes, keeps TENSORcnt cadence).
__device__ __forceinline__ void tdm_load_1d(u32 lds_off, u64 gaddr, u32 tile_elems) {
    v4u g0;
    g0.x = 1u;                                                // count=1, user desc
    g0.y = lds_off;                                           // LDS byte address
    g0.z = (u32)(gaddr & 0xFFFFFFFFull);                      // global_addr[31:0]
    g0.w = (u32)((gaddr >> 32) & 0x01FFFFFFull) | (2u << 30); // addr[56:32] | type=2
    v8i g1;
    g1.s0 = (int)(2u << 16);                 // data_size = 2 -> 4 bytes
    g1.s1 = (int)((u32)(NL * NL) << 16);     // tensor_dim0[15:0]  (bits 63:48)
    g1.s2 = (int)(1u << 16);                 // tensor_dim0 hi=0 | tensor_dim1=1
    g1.s3 = (int)(tile_elems << 16);         // tensor_dim1 hi=0 | tile_dim0
    g1.s4 = 0;                               // tile_dim1=0, tile_dim2=0 (unused)
    g1.s5 = (int)(NL * NL);                  // tensor_dim0_stride lo32
    g1.s6 = 0;
    g1.s7 = 0;
    v4i zz4 = {0, 0, 0, 0};                  // groups 2/3 unused (<=2D tensor)
    v8i zz8 = {0, 0, 0, 0, 0, 0, 0, 0};      // extra group (clang-23 6-arg form)
    __builtin_amdgcn_tensor_load_to_lds(g0, g1, zz4, zz4, zz8, 0);
}

__global__ __launch_bounds__(TPB) void crf_nll_kernel(
    const float* __restrict__ energy,   // [B, L, NL, NL]
    const int*   __restrict__ target,   // [B, L]
    const float* __restrict__ mask,     // [B, L]
    float*       __restrict__ out)      // [B]
{
    __shared__ float s_buf[DEPTH][NL * NL];  // streaming transition-matrix tiles
    __shared__ float s_part[NL];             // forward log-partition state
    __shared__ float s_m[ACT];               // per-partition running max
    __shared__ float s_s[ACT];               // per-partition exp-sum
    __shared__ float s_mask[NSTEP];
    __shared__ float s_red[TPB];
    __shared__ float s_tgt;

    const int b   = blockIdx.x;
    const int tid = threadIdx.x;
    const float* eb = energy + (size_t)b * NSTEP * NL * NL;

    // Stage mask row.
    for (int t = tid; t < NSTEP; t += TPB) s_mask[t] = mask[b * NSTEP + t];
    __syncthreads();

    // Target-path energy sum (fully parallel over t), deterministic reduction.
    float tacc = 0.0f;
    for (int t = tid; t < NSTEP; t += TPB) {
        int cur  = target[b * NSTEP + t];
        int prev = (t == 0) ? (NL - 1) : target[b * NSTEP + t - 1];
        tacc += eb[(size_t)t * NL * NL + (size_t)prev * NL + cur] * s_mask[t];
    }
    s_red[tid] = tacc;

    // part0 = energy[b,0,NL-1,:] * mask[b,0]
    if (tid < NL) s_part[tid] = eb[(NL - 1) * NL + tid] * s_mask[0];

    // Prime TDM pipeline: matrices for t = 1..DEPTH into slots 0..DEPTH-1.
    if (tid < 32) {
        for (int k = 0; k < DEPTH; ++k) {
            u32 loff = (u32)(uintptr_t)(const void*)&s_buf[k][0];
            tdm_load_1d(loff, (u64)(uintptr_t)(eb + (size_t)(1 + k) * NL * NL), NL * NL);
        }
    }
    __syncthreads();

    if (tid == 0) {   // deterministic serial sum (runs once, off critical path)
        float s = 0.0f;
        for (int k = 0; k < TPB; ++k) s += s_red[k];
        s_tgt = s;
    }

    const int  p = tid / NL;
    const int  j = tid % NL;
    const bool active = (tid < ACT);
    const float NEGINF = -__builtin_inff();

    for (int t = 1; t < NSTEP; ++t) {
        const int slot = (t - 1) % DEPTH;

        // Wave 0 owns TENSORcnt. In-order DMAs: at this point t-1+DEPTH have
        // been issued, so outstanding <= DEPTH-1 implies tile t has landed.
        if (tid < 32) __builtin_amdgcn_s_wait_tensorcnt(DEPTH - 1);
        __syncthreads();   // #1: tile ready; previous part update visible

        const float m_t = s_mask[t];
        if (active) {
            const float* ce = &s_buf[slot][0];
            float xs[17];
            float mx = NEGINF;
#pragma unroll
            for (int k = 0; k < 17; ++k) {
                int  i  = p + k * P;
                bool ok = (i < NL);
                int  ii = ok ? i : (NL - 1);
                float x = ok ? fmaf(ce[ii * NL + j], m_t, s_part[ii]) : NEGINF;
                xs[k] = x;
                mx = fmaxf(mx, x);
            }
            float ssum = 0.0f;
#pragma unroll
            for (int k = 0; k < 17; ++k) ssum += fast_exp(xs[k] - mx);
            s_m[tid] = mx;
            s_s[tid] = ssum;
        }
        __syncthreads();   // #2: partials written; tile fully consumed

        // Refill this slot immediately (overlaps the DMA with the combine).
        // Dummy NOP descriptor at the tail keeps the wait immediate constant.
        if (tid < 32) {
            int tn = t + DEPTH;
            u32 loff = (u32)(uintptr_t)(const void*)&s_buf[slot][0];
            if (tn < NSTEP)
                tdm_load_1d(loff, (u64)(uintptr_t)(eb + (size_t)tn * NL * NL), NL * NL);
            else
                tdm_load_1d(loff, (u64)(uintptr_t)eb, 0u);
        }

        // Combine the P partial (max,sum) pairs per column; blend with mask.
        // Next iteration's barrier #1 publishes s_part before it is re-read.
        if (tid < NL) {
            float mx = s_m[tid], ssum = s_s[tid];
#pragma unroll
            for (int q = 1; q < P; ++q) {
                float m2 = s_m[q * NL + tid], s2 = s_s[q * NL + tid];
                float mn = fmaxf(mx, m2);
                ssum = ssum * fast_exp(mx - mn) + s2 * fast_exp(m2 - mn);
                mx = mn;
            }
            float pnew = mx + fast_log(ssum);
            float po   = s_part[tid];
            s_part[tid] = fmaf(pnew - po, m_t, po);   // part += (pnew-part)*m
        }
    }

    __syncthreads();
    if (tid == 0) {
        float mx = s_part[0];
        for (int i = 1; i < NL; ++i) mx = fmaxf(mx, s_part[i]);
        float ssum = 0.0f;
        for (int i = 0; i < NL; ++i) ssum += fast_exp(s_part[i] - mx);
        out[b] = (mx + fast_log(ssum)) - s_tgt;
    }
}

extern "C" void kernel_launch(void* const* d_in, const int* in_sizes, int n_in,
                              void* d_out, int out_size, void* d_ws, size_t ws_size,
                              hipStream_t stream) {
    const float* energy = (const float*)d_in[0];
    const int*   target = (const int*)d_in[1];
    const float* maskp  = (const float*)d_in[2];
    float*       outp   = (float*)d_out;
    (void)in_sizes; (void)n_in; (void)out_size; (void)d_ws; (void)ws_size;
    crf_nll_kernel<<<NBATCH, TPB, 0, stream>>>(energy, target, maskp, outp);
}